// QuantumAttentionOld_25941602468030
// MI455X (gfx1250) — compile-verified
//
#include <hip/hip_runtime.h>
#include <hip/hip_bf16.h>
#include <stdint.h>

typedef __attribute__((ext_vector_type(2))) float v2f;
typedef __attribute__((ext_vector_type(4))) float v4f;
typedef __attribute__((ext_vector_type(8))) float v8f;

#define B_    2
#define L_    2048
#define H_    16
#define E_    128
#define S_    2048
#define KPAD  132   // LDS row stride for K/V tiles (floats); 132 % 64 = 4 -> conflict-free
#define PPAD  18    // LDS row stride for P scratch; 18m % 64 distinct for m=0..15

__global__ __launch_bounds__(128, 1) void qattn_flash(
    const float* __restrict__ keys,
    const float* __restrict__ values,
    float* __restrict__ out)
{
    // double-buffered K/V tiles + per-wave P transpose scratch (~38 KB of 320 KB)
    __shared__ alignas(16) float kt[2][16 * KPAD];
    __shared__ alignas(16) float vt[2][16 * KPAD];
    __shared__ alignas(16) float pt[4][16 * PPAD];

    const int tid  = threadIdx.x;
    const int wave = tid >> 5;
    const int lane = tid & 31;
    const int hi   = lane >> 4;   // which 16-lane half
    const int lp   = lane & 15;

    const int blk  = blockIdx.x;
    const int mblk = blk & 31;            // 32 blocks of 64 query rows
    const int h    = (blk >> 5) & 15;
    const int b    = blk >> 9;

    const int R0    = mblk * 64;
    const int rowlo = R0 + wave * 16;     // this wave's 16 query rows

    // ---- "Q" (= values rows) A-fragments, resident in registers ----
    // f32 16x4 A layout: lanes 0-15 hold K=e0,e0+1; lanes 16-31 hold e0+2,e0+3.
    v2f qf[32];
    {
        const float* qbase = values + (((size_t)b * L_ + (rowlo + lp)) * H_ + h) * E_;
        #pragma unroll
        for (int i = 0; i < 32; ++i)
            qf[i] = *(const v2f*)(qbase + 4 * i + 2 * hi);
    }

    v8f o[8];
    #pragma unroll
    for (int n = 0; n < 8; ++n) o[n] = {};
    float rm[8], rl[8];
    #pragma unroll
    for (int g = 0; g < 8; ++g) { rm[g] = -INFINITY; rl[g] = 0.0f; }

    const float cst    = 0.044194173824159216f;   // ENT_FACTOR / sqrt(E) = 0.5/sqrt(128)
    const int   ntiles = (R0 >> 4) + 4;           // causal: KV tiles up to row R0+63

    // --- async DMA of one 16x128 K tile + V tile straight into LDS (ASYNCcnt) ---
    auto issue_tile = [&](int t, int sel) {
        const uint32_t kb = (uint32_t)(uintptr_t)&kt[sel][0];
        const uint32_t vb = (uint32_t)(uintptr_t)&vt[sel][0];
        #pragma unroll
        for (int it = 0; it < 4; ++it) {
            const int i  = tid + it * 128;        // 512 float4 per tile
            const int r  = i >> 5;
            const int c4 = (i & 31) << 2;
            const size_t  goff = (((size_t)b * S_ + (t << 4) + r) * H_ + h) * E_ + c4;
            const uint32_t ldsoff = (uint32_t)(r * KPAD + c4) * 4u;
            uint64_t ksrc = (uint64_t)(uintptr_t)(keys   + goff);
            uint64_t vsrc = (uint64_t)(uintptr_t)(values + goff);
            uint32_t kdst = kb + ldsoff;
            uint32_t vdst = vb + ldsoff;
            asm volatile("global_load_async_to_lds_b128 %0, %1, off"
                         :: "v"(kdst), "v"(ksrc) : "memory");
            asm volatile("global_load_async_to_lds_b128 %0, %1, off"
                         :: "v"(vdst), "v"(vsrc) : "memory");
        }
    };

    issue_tile(0, 0);

    for (int t = 0; t < ntiles; ++t) {
        const int cur = t & 1;
        const int s0  = t << 4;

        asm volatile("s_wait_asynccnt 0x0" ::: "memory");  // my DMA writes landed in LDS
        __syncthreads();                                   // everyone's tile t is visible

        if (t + 1 < ntiles) issue_tile(t + 1, cur ^ 1);    // prefetch next tile during compute

        // ---- S = Vq * K^T : 32 x v_wmma_f32_16x16x4_f32 (B frags chunk-preloaded) ----
        v8f c = {};
        #pragma unroll
        for (int i0 = 0; i0 < 32; i0 += 8) {
            v2f bf[8];
            #pragma unroll
            for (int j = 0; j < 8; ++j)
                bf[j] = *(const v2f*)&kt[cur][lp * KPAD + 4 * (i0 + j) + 2 * hi];
            #pragma unroll
            for (int j = 0; j < 8; ++j)
                c = __builtin_amdgcn_wmma_f32_16x16x4_f32(false, qf[i0 + j], false, bf[j],
                                                          (short)0, c, false, false);
        }

        // ---- scale + causal mask + online softmax (C layout: row g+8*hi, col lp) ----
        float alpha[8], p[8];
        const int col = s0 + lp;
        #pragma unroll
        for (int g = 0; g < 8; ++g) {
            const int row = rowlo + g + 8 * hi;
            float x = c[g] * cst;
            if (col > row) x = -INFINITY;
            float mx = x;
            #pragma unroll
            for (int d = 1; d < 16; d <<= 1)
                mx = fmaxf(mx, __shfl_xor(mx, d, 16));
            const float mnew = fmaxf(rm[g], mx);
            alpha[g] = __expf(rm[g] - mnew);     // rm=-inf only before first tile -> 0
            const float pe = __expf(x - mnew);   // x=-inf -> 0
            float sum = pe;
            #pragma unroll
            for (int d = 1; d < 16; d <<= 1)
                sum += __shfl_xor(sum, d, 16);
            rl[g] = rl[g] * alpha[g] + sum;
            rm[g] = mnew;
            p[g]  = pe;
        }
        #pragma unroll
        for (int n = 0; n < 8; ++n)
            #pragma unroll
            for (int g = 0; g < 8; ++g)
                o[n][g] *= alpha[g];

        // stage P through LDS: C-layout -> A-layout transpose (wave-private scratch)
        #pragma unroll
        for (int g = 0; g < 8; ++g)
            pt[wave][(g + 8 * hi) * PPAD + lp] = p[g];
        __syncthreads();

        // ---- O += P * V : 8 N-tiles x 4 K-steps of v_wmma_f32_16x16x4_f32 ----
        #pragma unroll
        for (int n = 0; n < 8; ++n) {
            #pragma unroll
            for (int k0 = 0; k0 < 16; k0 += 4) {
                v2f a = *(const v2f*)&pt[wave][lp * PPAD + k0 + 2 * hi];
                const int srow = k0 + 2 * hi;
                v2f bb;
                bb.x = vt[cur][srow * KPAD + n * 16 + lp];
                bb.y = vt[cur][(srow + 1) * KPAD + n * 16 + lp];
                o[n] = __builtin_amdgcn_wmma_f32_16x16x4_f32(false, a, false, bb,
                                                             (short)0, o[n], false, false);
            }
        }
    }

    // ---- normalize and write (coalesced 64B per 16-lane half) ----
    #pragma unroll
    for (int g = 0; g < 8; ++g) rl[g] = 1.0f / rl[g];
    #pragma unroll
    for (int n = 0; n < 8; ++n) {
        #pragma unroll
        for (int g = 0; g < 8; ++g) {
            const int row = rowlo + g + 8 * hi;
            out[(((size_t)b * L_ + row) * H_ + h) * E_ + n * 16 + lp] = o[n][g] * rl[g];
        }
    }
}

extern "C" void kernel_launch(void* const* d_in, const int* in_sizes, int n_in,
                              void* d_out, int out_size, void* d_ws, size_t ws_size,
                              hipStream_t stream) {
    (void)in_sizes; (void)n_in; (void)out_size; (void)d_ws; (void)ws_size;
    // d_in[0] = queries (unused by the math: classical QK scores are never added)
    // d_in[3] = q_params (unused: constant shift cancels in softmax)
    const float* keys   = (const float*)d_in[1];
    const float* values = (const float*)d_in[2];
    float* out = (float*)d_out;

    const int blocks = B_ * H_ * (L_ / 64);   // 1024
    qattn_flash<<<blocks, 128, 0, stream>>>(keys, values, out);
}